// neuron_layer_hardreset_8469675507795
// MI455X (gfx1250) — compile-verified
//
#include <hip/hip_runtime.h>

typedef __attribute__((ext_vector_type(2))) float v2f;
typedef __attribute__((ext_vector_type(8))) float v8f;

#define B_SZ 128
#define D_SZ 2048
#define N_SZ 2048
#define T_SZ 64
#define THRESH 0.5f

#define KC   64   // d-chunk staged per iteration
#define NBLK 128  // neurons per block (8 waves x 16)
#define AS   72   // a_lds row stride (rows = d, cols = t)  -> conflict-free frag reads
#define WS   68   // w_lds row stride (rows = n, cols = d)  -> conflict-free frag reads

// D = A(16x4, M=t, K=d) x B(4x16, K=d, N=n) + C  in fp32
__device__ __forceinline__ v8f wmma_f32(v2f a, v2f b, v8f c) {
  return __builtin_amdgcn_wmma_f32_16x16x4_f32(false, a, false, b, (short)0, c,
                                               false, false);
}

__global__ __launch_bounds__(256, 2) void lif_fused_wmma_kernel(
    const float* __restrict__ spikes,  // [B, D, T]
    const float* __restrict__ W,       // [N, D]
    const float* __restrict__ bias,    // [N]
    const float* __restrict__ decay,   // [N]
    float* __restrict__ out)           // spikes [B,N,T] then v_final [B,N]
{
  __shared__ float a_lds[KC * AS];    // 18.4 KB
  __shared__ float w_lds[NBLK * WS];  // 34.8 KB

  const int tid    = threadIdx.x;
  const int wave   = tid >> 5;
  const int lane   = tid & 31;
  const int lane_n = lane & 15;
  const int hi     = lane >> 4;  // half-wave: selects K offset +2 / M offset +8

  const int b  = blockIdx.y;
  const int n0 = blockIdx.x * NBLK;
  const int n  = n0 + wave * 16 + lane_n;

  const float* gA0 = spikes + (size_t)b * (D_SZ * T_SZ);
  const float* gW0 = W + (size_t)n0 * D_SZ;

  v8f c0 = {}, c1 = {}, c2 = {}, c3 = {};  // t-groups 0..3 (16 t each) x 16 n

  for (int kc = 0; kc < D_SZ / KC; ++kc) {
    const float* gA = gA0 + kc * (KC * T_SZ);  // contiguous 64d x 64t block
    const float* gW = gW0 + kc * KC;

    __syncthreads();
    // Stage A chunk: spikes[b, kc*64 + d, t] -> a_lds[d*AS + t]
#pragma unroll
    for (int i = 0; i < 4; ++i) {
      int idx = (tid + i * 256) * 4;
      int d = idx >> 6;
      int t = idx & 63;
      *(float4*)&a_lds[d * AS + t] = *(const float4*)(gA + idx);
    }
    // Stage W tile: W[n0+nn, kc*64 + dd] -> w_lds[nn*WS + dd]
#pragma unroll
    for (int i = 0; i < 8; ++i) {
      int idx = tid + i * 256;
      int nn = idx >> 4;
      int dd = (idx & 15) * 4;
      *(float4*)&w_lds[nn * WS + dd] = *(const float4*)(gW + (size_t)nn * D_SZ + dd);
    }
    __syncthreads();

    if (kc + 1 < D_SZ / KC) {  // hint next chunk into cache (global_prefetch_b8)
      __builtin_prefetch(gA + KC * T_SZ + tid * 16, 0, 1);
      __builtin_prefetch(gW + KC + (size_t)(tid & 127) * D_SZ + (tid >> 7) * 32, 0, 1);
    }

    // B fragment row for this wave's 16 neurons; K = k0 + 2*hi + {0,1}
    const float* wrow  = &w_lds[(wave * 16 + lane_n) * WS + 2 * hi];
    // A fragment base; element (k0 + 2*hi + r, tg*16 + lane_n)
    const float* abase = &a_lds[(2 * hi) * AS + lane_n];

#pragma unroll
    for (int k0 = 0; k0 < KC; k0 += 4) {
      float2 bb = *(const float2*)(wrow + k0);  // ds_load_b64
      v2f Bv; Bv.x = bb.x; Bv.y = bb.y;
      const float* ak = abase + k0 * AS;
      v2f A0, A1, A2, A3;
      A0.x = ak[0];       A0.y = ak[AS + 0];
      A1.x = ak[16];      A1.y = ak[AS + 16];
      A2.x = ak[32];      A2.y = ak[AS + 32];
      A3.x = ak[48];      A3.y = ak[AS + 48];
      c0 = wmma_f32(A0, Bv, c0);
      c1 = wmma_f32(A1, Bv, c1);
      c2 = wmma_f32(A2, Bv, c2);
      c3 = wmma_f32(A3, Bv, c3);
    }
  }

  const float bn = bias[n];
  const float dn = decay[n];

  // Gather full time series per lane: lane l and l^16 hold the same n;
  // C layout: reg r -> t = tg*16 + r + 8*hi.  One shfl_xor(16) completes it.
  float p[T_SZ];
#pragma unroll
  for (int tg = 0; tg < 4; ++tg) {
    v8f cr = (tg == 0) ? c0 : (tg == 1) ? c1 : (tg == 2) ? c2 : c3;
#pragma unroll
    for (int r = 0; r < 8; ++r) {
      float own = cr[r] + bn;                // psp = GEMM + bias
      float oth = __shfl_xor(own, 16, 32);
      p[tg * 16 + r]     = hi ? oth : own;   // t = tg*16 + r
      p[tg * 16 + 8 + r] = hi ? own : oth;   // t = tg*16 + 8 + r
    }
  }

  // LIF scan with hard reset: v = (1-s)*v*decay + psp_t ; s = v > 0.5
  float v = 0.f, s = 0.f;
#pragma unroll
  for (int t = 0; t < T_SZ; ++t) {
    v = (1.f - s) * v * dn + p[t];
    s = (v > THRESH) ? 1.f : 0.f;
    p[t] = s;  // reuse registers for spike output
  }

  // out spikes [B,N,T]: t-contiguous per (b,n) -> 128B bursts per lane.
  float* outS = out + ((size_t)b * N_SZ + n) * T_SZ;
  if (hi == 0) {
#pragma unroll
    for (int j = 0; j < 8; ++j) {
      float4 q = make_float4(p[4 * j], p[4 * j + 1], p[4 * j + 2], p[4 * j + 3]);
      *(float4*)(outS + 4 * j) = q;
    }
    out[(size_t)B_SZ * N_SZ * T_SZ + (size_t)b * N_SZ + n] = v;  // v_final
  } else {
#pragma unroll
    for (int j = 0; j < 8; ++j) {
      float4 q = make_float4(p[32 + 4 * j], p[33 + 4 * j],
                             p[34 + 4 * j], p[35 + 4 * j]);
      *(float4*)(outS + 32 + 4 * j) = q;
    }
  }
}

extern "C" void kernel_launch(void* const* d_in, const int* in_sizes, int n_in,
                              void* d_out, int out_size, void* d_ws, size_t ws_size,
                              hipStream_t stream) {
  (void)in_sizes; (void)n_in; (void)out_size; (void)d_ws; (void)ws_size;
  const float* spikes = (const float*)d_in[0];
  const float* W      = (const float*)d_in[1];
  const float* bias   = (const float*)d_in[2];
  const float* decay  = (const float*)d_in[3];
  float* out          = (float*)d_out;

  dim3 grid(N_SZ / NBLK, B_SZ);  // 16 x 128 = 2048 blocks
  lif_fused_wmma_kernel<<<grid, 256, 0, stream>>>(spikes, W, bias, decay, out);
}